// Tril2Diff_21706764714000
// MI455X (gfx1250) — compile-verified
//
#include <hip/hip_runtime.h>
#include <math.h>

typedef __attribute__((ext_vector_type(2))) float v2f;
typedef __attribute__((ext_vector_type(8))) float v8f;

#define DIM     128
#define NTRIL   8256          // 128*129/2
#define LSTRIDE 132           // 128 + 4 pad: bank-conflict-free fragment loads
#define NWAVES  8
#define BLOCK   (NWAVES * 32) // wave32

// Serpentine deal of the 64 output tiles (sorted by K-cost = min(I,J)+1,
// dealt boustrophedon across 8 waves). Per-wave K-tile totals:
// {28,26,26,26,24,24,24,26} vs 20..36 for the naive (i+wave)&7 mapping.
// Entry = I*8 + J; kTileMap[wave*8 + i].
__device__ const unsigned char kTileMap[64] = {
    63, 39, 27, 21, 50, 57, 15,  7,   // wave 0
    54, 60, 35, 42, 22, 14,  0, 56,   // wave 1
    62, 38, 28, 20, 58, 49,  8,  6,   // wave 2
    55, 52, 43, 34, 23, 13,  1, 48,   // wave 3
    45, 37, 29, 19,  9, 41, 16,  5,   // wave 4
    53, 44, 51, 26, 17, 12,  2, 40,   // wave 5
    46, 36, 30, 18, 10, 33, 24,  4,   // wave 6
    61, 47, 59, 31, 25, 11,  3, 32    // wave 7
};

__global__ __launch_bounds__(BLOCK)
void tril2diff_kernel(const float* __restrict__ x, float* __restrict__ out, int batch) {
    extern __shared__ float Ls[]; // [DIM][LSTRIDE]
    const int b = blockIdx.x;
    if (b >= batch) return;
    const int tid = threadIdx.x;

    // ---- zero L staging area ----
    for (int i = tid; i < DIM * LSTRIDE; i += BLOCK) Ls[i] = 0.0f;
    __syncthreads();

    // ---- scatter x -> lower-triangular L in LDS; softplus the diagonal ----
    // x is streamed exactly once across the dispatch -> non-temporal loads.
    const float* xb = x + (size_t)b * NTRIL;
    for (int p = tid; p < NTRIL; p += BLOCK) {
        float v = __builtin_nontemporal_load(xb + p);
        int r = (int)((sqrtf(8.0f * (float)p + 1.0f) - 1.0f) * 0.5f);
        while (((r + 1) * (r + 2)) / 2 <= p) ++r;   // fixup for float rounding
        while ((r * (r + 1)) / 2 > p) --r;
        const int c = p - (r * (r + 1)) / 2;
        if (c == r) v = log1pf(expf(v) + 1e-12f);   // diag = log1p(exp(x)+1e-12)
        Ls[r * LSTRIDE + c] = v;
    }
    __syncthreads();

    // ---- C = L * L^T via V_WMMA_F32_16X16X4_F32, triangular K pruning ----
    const int lane = tid & 31;
    const int wave = __builtin_amdgcn_readfirstlane(tid >> 5); // force wave-uniform
    const int rsel = lane & 15;        // row within 16-tile (A) / col within tile (B)
    const int kb   = (lane >> 4) * 2;  // K pair select per ISA 32-bit A 16x4 layout

    float* outb = out + (size_t)b * (DIM * DIM);

    for (int i = 0; i < 8; ++i) {
        const int t  = kTileMap[wave * 8 + i];   // wave-uniform scalar load
        const int I  = t >> 3;
        const int J  = t & 7;
        const int KT = (I < J ? I : J) + 1;      // K-tiles 0..min(I,J): rest is zeros

        const float* __restrict__ arow = &Ls[(I * 16 + rsel) * LSTRIDE];
        const float* __restrict__ brow = &Ls[(J * 16 + rsel) * LSTRIDE]; // B = (L tile J)^T

        v8f acc = {};
        for (int kt = 0; kt < KT; ++kt) {
            const int kbase = kt * 16 + kb;
#pragma unroll
            for (int ks = 0; ks < 4; ++ks) {
                const int k0 = kbase + ks * 4;
                v2f a;  a.x  = arow[k0]; a.y  = arow[k0 + 1];
                v2f bb; bb.x = brow[k0]; bb.y = brow[k0 + 1];
                acc = __builtin_amdgcn_wmma_f32_16x16x4_f32(
                    /*neg_a=*/false, a, /*neg_b=*/false, bb,
                    /*c_mod=*/(short)0, acc, /*reuse_a=*/false, /*reuse_b=*/false);
            }
        }

        // D layout: VGPR r -> rows {r, r+8} selected by lane half; N = lane&15.
        // Output is write-once streaming (512 MB >> L2) -> non-temporal stores.
        float* op = outb + (size_t)(I * 16 + (lane >> 4) * 8) * DIM + (J * 16 + rsel);
#pragma unroll
        for (int rr = 0; rr < 8; ++rr)
            __builtin_nontemporal_store(acc[rr], op + rr * DIM);
    }
}

extern "C" void kernel_launch(void* const* d_in, const int* in_sizes, int n_in,
                              void* d_out, int out_size, void* d_ws, size_t ws_size,
                              hipStream_t stream) {
    const float* x = (const float*)d_in[0];
    float* out = (float*)d_out;
    const int batch = in_sizes[0] / NTRIL;
    const size_t shmem = (size_t)DIM * LSTRIDE * sizeof(float);
    tril2diff_kernel<<<dim3(batch), dim3(BLOCK), shmem, stream>>>(x, out, batch);
}